// GVAE_9397388444087
// MI455X (gfx1250) — compile-verified
//
#include <hip/hip_runtime.h>
#include <hip/hip_bf16.h>

// ---------------------------------------------------------------------------
// GVAE forward for MI455X (gfx1250, wave32).
//  * All matmuls via V_WMMA_F32_16X16X4_F32 (fp32-exact, wave-level 16x16 tile).
//  * Attention FiLM + ea + softmax(d) + na*v + the entire edge MLP
//    (We -> Me1 -> Me2) fused in one kernel; ea lives only in LDS (32 KB).
//  * GEMM K-loop is fully strength-reduced: no 64-bit multiplies, no
//    EXEC-masked loads on the (common) full-tile path.
//  * Input pointer order assumed = jax.tree_util.tree_leaves(setup_inputs())
//    (dict keys sorted at every level):
//      0 edges, 1 eps, 2 nodes,
//      dec: 3 ce1, 4 ce2, 5 ceb1, 6 ceb2, 7 cn1, 8 cn2, 9 cnb1, 10 cnb2,
//           11 oe1, 12 oe2, 13 oeb1, 14 oeb2, 15 on1, 16 on2, 17 onb1, 18 onb2,
//           19..40 tf leaves (Fb1,Fb2,Fw1,Fw2,Me1,Me2,Mn1,Mn2,We,Wk,Wn,Wq,Wv,
//                             be,bk,bn,bq,bv,me1,me2,mn1,mn2)
//      enc: 41 Wle, 42 ble, 43 inE1, 44 inE2, 45 inEb1, 46 inEb2,
//           47 inN1, 48 inN2, 49 inNb1, 50 inNb2, 51 m1, 52 m2, 53 mb1, 54 mb2,
//           55..76 tf leaves, 77 v1, 78 v2, 79 vb1, 80 vb2
// ---------------------------------------------------------------------------

typedef __attribute__((ext_vector_type(2))) float v2f;
typedef __attribute__((ext_vector_type(8))) float v8f;

__device__ __forceinline__ v8f wmma4(v2f a, v2f b, v8f c) {
#if defined(__gfx1250__) && __has_builtin(__builtin_amdgcn_wmma_f32_16x16x4_f32)
  // 8 args: (neg_a, A, neg_b, B, c_mod, C, reuse_a, reuse_b)
  return __builtin_amdgcn_wmma_f32_16x16x4_f32(false, a, false, b, (short)0, c,
                                               false, false);
#else
  c[0] += a.x * b.x;  // host-pass / non-gfx1250 stub (never used on device)
  return c;
#endif
}

// ---------------------------------------------------------------------------
// Generic WMMA GEMM: C[M,N] = act(A @ W[K,N] + bias).
// A-element address:  (m / aRowsPB)*aBatch + (m % aRowsPB)*aRowStride
//                   + (kchunk)*aHead + (k in chunk)     [64-wide K chunks]
//   plain [M,K]   : aRowsPB=huge, aRowStride=K, aHead=64
//   head-concat wv: aRowsPB=16,  aBatch=H*16*64, aRowStride=64, aHead=16*64
// One wave per 16x16 output tile, 4 waves per block.
// act: 0=none, 1=relu, 2=sigmoid on column 0 only.
// FULL=true  : N % 16 == 0 and M % 16 == 0 -> no guards anywhere.
// ---------------------------------------------------------------------------
template <bool FULL>
__global__ void __launch_bounds__(128)
gemm_kernel(const float* __restrict__ A, const float* __restrict__ W,
            const float* __restrict__ bias, float* __restrict__ C,
            int M, int N, int K,
            int aRowsPB, long long aBatch, int aRowStride, int aHead, int act) {
  const int wave = threadIdx.x >> 5, lane = threadIdx.x & 31;
  const int Nt = (N + 15) >> 4;
  const int nt = blockIdx.x * 4 + wave;
  if (nt >= Nt) return;  // wave-uniform
  const int m0 = blockIdx.y << 4;
  const int n0 = nt << 4;
  const int half = lane >> 4, l = lane & 15;
  int mRow = m0 + l;
  if (!FULL && mRow >= M) mRow = M - 1;
  const long long aBase = (long long)(mRow / aRowsPB) * aBatch +
                          (long long)(mRow % aRowsPB) * (long long)aRowStride;
  const int col = n0 + l;
  const bool colOk = FULL || (col < N);
  const int colC = colOk ? col : (N - 1);

  // strength-reduced pointers: ka = kc + kk + 2*half (kk+2*half < 64 always)
  const float* ap = A + aBase + (half << 1);
  const float* wp = W + (long long)(half << 1) * N + colC;
  const long long wStep = 4LL * N;

  v8f c = {0.f, 0.f, 0.f, 0.f, 0.f, 0.f, 0.f, 0.f};
  for (int kc = 0; kc < K; kc += 64) {
    const float* a2 = ap;
    const float* w2 = wp;
    const int kmax = (K - kc < 64) ? (K - kc) : 64;
    for (int kk = 0; kk < kmax; kk += 4) {
      v2f a, b;
      a.x = a2[0];
      a.y = a2[1];
      if (FULL) {
        b.x = w2[0];
        b.y = w2[N];
      } else {
        b.x = colOk ? w2[0] : 0.f;
        b.y = colOk ? w2[N] : 0.f;
      }
      c = wmma4(a, b, c);
      a2 += 4;
      w2 += wStep;
    }
    ap += aHead;
    wp += (long long)64 * N;
  }

  if (!FULL && !colOk) return;
  const float bb = bias ? bias[col] : 0.f;
  float* cp = C + (long long)(m0 + (half << 3)) * N + col;
#pragma unroll
  for (int r = 0; r < 8; ++r) {
    if (FULL || (m0 + r + (half << 3)) < M) {
      float x = c[r] + bb;
      if (act == 1)
        x = fmaxf(x, 0.f);
      else if (act == 2 && col == 0)
        x = 1.f / (1.f + __expf(-x));
      cp[(long long)r * N] = x;
    }
  }
}

// ---------------------------------------------------------------------------
// QKV projection: grid.z = (which in {q,k,v}) * 8 + head.
// nodes [B*16, 64] @ W_h[64,64] + b_h -> out[b,h,n,d].  blockIdx.y = b.
// ---------------------------------------------------------------------------
__global__ void __launch_bounds__(128)
qkv_kernel(const float* __restrict__ nodes, const float* __restrict__ Wq,
           const float* __restrict__ Wk, const float* __restrict__ Wv,
           const float* __restrict__ bq, const float* __restrict__ bk,
           const float* __restrict__ bvp, float* __restrict__ q,
           float* __restrict__ k, float* __restrict__ v) {
  const int wave = threadIdx.x >> 5, lane = threadIdx.x & 31;
  const int z = blockIdx.z, which = z >> 3, h = z & 7;
  const float* W = (which == 0 ? Wq : which == 1 ? Wk : Wv) + h * 4096;
  const float* bias = (which == 0 ? bq : which == 1 ? bk : bvp) + h * 64;
  float* out = (which == 0 ? q : which == 1 ? k : v);
  const int b = blockIdx.y;
  const int m0 = b << 4, n0 = wave << 4;
  const int half = lane >> 4, l = lane & 15;
  const float* ap = nodes + (m0 + l) * 64 + (half << 1);
  const float* wp = W + (half << 1) * 64 + n0 + l;
  v8f c = {0.f, 0.f, 0.f, 0.f, 0.f, 0.f, 0.f, 0.f};
#pragma unroll
  for (int kk = 0; kk < 64; kk += 4) {
    v2f a, bf;
    a.x = ap[0];
    a.y = ap[1];
    bf.x = wp[0];
    bf.y = wp[64];
    c = wmma4(a, bf, c);
    ap += 4;
    wp += 256;
  }
  const float bb = bias[n0 + l];
  float* obase = out + ((b * 8 + h) * 16) * 64;
#pragma unroll
  for (int r = 0; r < 8; ++r) {
    const int n = r + (half << 3);
    obase[n * 64 + n0 + l] = c[r] + bb;
  }
}

// ---------------------------------------------------------------------------
// Fused attention + edge MLP.  One block per (b, i); 512 threads = 16 waves.
//  - FiLM (edges @ Fw1/Fw2) computed in-register (weights L2-resident),
//  - ea kept in LDS (8*16*64 fp32 = 32 KB), never written to HBM,
//  - softmax over d (64) via wave32 shfl_xor reductions,
//  - wv j-reduction via LDS ds_add_f32 atomics,
//  - ea(16x512) @ We -> relu -> @Me1 -> relu -> @Me2 done with WMMA in-block.
// ---------------------------------------------------------------------------
__global__ void __launch_bounds__(512)
attn_kernel(const float* __restrict__ q, const float* __restrict__ k,
            const float* __restrict__ v, const float* __restrict__ edges,
            const float* __restrict__ Fw1, const float* __restrict__ Fw2,
            const float* __restrict__ Fb1, const float* __restrict__ Fb2,
            const float* __restrict__ We, const float* __restrict__ be,
            const float* __restrict__ Me1, const float* __restrict__ me1,
            const float* __restrict__ Me2, const float* __restrict__ me2,
            float* __restrict__ wvOut, float* __restrict__ eOut) {
  __shared__ float sE[512];    // edges[j][e]   16x32
  __shared__ float sQ[512];    // q_i[h][d]     8x64
  __shared__ float sEA[8192];  // ea[h][j][d]   8x16x64
  __shared__ float sWV[512];   // wv[h][d]
  __shared__ float sT0[512];   // 16x32 stage temp
  __shared__ float sT1[512];
  const int tid = threadIdx.x, wave = tid >> 5, lane = tid & 31;
  const int b = blockIdx.x >> 4, i = blockIdx.x & 15;

  sE[tid] = edges[(b * 256 + i * 16) * 32 + tid];
  sQ[tid] = q[((b * 8 + (tid >> 6)) * 16 + i) * 64 + (tid & 63)];
  sWV[tid] = 0.f;
  __syncthreads();

  // 128 (h,j) pairs; each wave handles 8 of them; lane covers d and d+32.
  for (int it = 0; it < 8; ++it) {
    const int p = it * 16 + wave;
    const int h = p >> 4, j = p & 15;
    const int d0 = lane, d1 = lane + 32;
    const float* krow = k + ((b * 8 + h) * 16 + j) * 64;
    const float att0 = sQ[h * 64 + d0] * krow[d0] * 0.125f;
    const float att1 = sQ[h * 64 + d1] * krow[d1] * 0.125f;
    float mul0 = Fb1[h * 64 + d0], mul1 = Fb1[h * 64 + d1];
    float add0 = Fb2[h * 64 + d0], add1 = Fb2[h * 64 + d1];
    const float* fw1 = Fw1 + h * 2048 + d0;
    const float* fw2 = Fw2 + h * 2048 + d0;
#pragma unroll 8
    for (int e = 0; e < 32; ++e) {
      const float ev = sE[j * 32 + e];  // LDS broadcast
      mul0 = fmaf(ev, fw1[0], mul0);
      mul1 = fmaf(ev, fw1[32], mul1);
      add0 = fmaf(ev, fw2[0], add0);
      add1 = fmaf(ev, fw2[32], add1);
      fw1 += 64;
      fw2 += 64;
    }
    const float ea0 = mul0 * att0 + add0 + att0;
    const float ea1 = mul1 * att1 + add1 + att1;
    sEA[h * 1024 + j * 64 + d0] = ea0;
    sEA[h * 1024 + j * 64 + d1] = ea1;
    // softmax over the 64 d-values (2 per lane, wave32 reduction)
    float m = fmaxf(ea0, ea1);
    for (int off = 16; off > 0; off >>= 1) m = fmaxf(m, __shfl_xor(m, off, 32));
    const float x0 = __expf(ea0 - m), x1 = __expf(ea1 - m);
    float ssum = x0 + x1;
    for (int off = 16; off > 0; off >>= 1) ssum += __shfl_xor(ssum, off, 32);
    const float inv = 1.f / ssum;
    const float* vrow = v + ((b * 8 + h) * 16 + j) * 64;
    atomicAdd(&sWV[h * 64 + d0], x0 * inv * vrow[d0]);  // ds_add_f32
    atomicAdd(&sWV[h * 64 + d1], x1 * inv * vrow[d1]);
  }
  __syncthreads();
  wvOut[((b * 8 + (tid >> 6)) * 16 + i) * 64 + (tid & 63)] = sWV[tid];

  const int half = lane >> 4, l = lane & 15;
  // stage 1: t0 = relu(ea(16x512) @ We(512x32) + be)   (waves 0,1)
  if (wave < 2) {
    const int n0 = wave << 4;
    const float* wp = We + (half << 1) * 32 + n0 + l;
    v8f c = {0.f, 0.f, 0.f, 0.f, 0.f, 0.f, 0.f, 0.f};
    for (int kk = 0; kk < 512; kk += 4) {
      const int ka = kk + (half << 1);
      v2f a, bf;
      a.x = sEA[(ka >> 6) * 1024 + l * 64 + (ka & 63)];
      a.y = sEA[(ka >> 6) * 1024 + l * 64 + (ka & 63) + 1];
      bf.x = wp[0];
      bf.y = wp[32];
      c = wmma4(a, bf, c);
      wp += 128;
    }
    const float bb = be[n0 + l];
#pragma unroll
    for (int r = 0; r < 8; ++r)
      sT0[(r + (half << 3)) * 32 + n0 + l] = fmaxf(c[r] + bb, 0.f);
  }
  __syncthreads();
  // stage 2: t1 = relu(t0 @ Me1 + me1)
  if (wave < 2) {
    const int n0 = wave << 4;
    const float* wp = Me1 + (half << 1) * 32 + n0 + l;
    v8f c = {0.f, 0.f, 0.f, 0.f, 0.f, 0.f, 0.f, 0.f};
#pragma unroll
    for (int kk = 0; kk < 32; kk += 4) {
      const int ka = kk + (half << 1);
      v2f a, bf;
      a.x = sT0[l * 32 + ka];
      a.y = sT0[l * 32 + ka + 1];
      bf.x = wp[0];
      bf.y = wp[32];
      c = wmma4(a, bf, c);
      wp += 128;
    }
    const float bb = me1[n0 + l];
#pragma unroll
    for (int r = 0; r < 8; ++r)
      sT1[(r + (half << 3)) * 32 + n0 + l] = fmaxf(c[r] + bb, 0.f);
  }
  __syncthreads();
  // stage 3: e_out = t1 @ Me2 + me2  (no relu)
  if (wave < 2) {
    const int n0 = wave << 4;
    const float* wp = Me2 + (half << 1) * 32 + n0 + l;
    v8f c = {0.f, 0.f, 0.f, 0.f, 0.f, 0.f, 0.f, 0.f};
#pragma unroll
    for (int kk = 0; kk < 32; kk += 4) {
      const int ka = kk + (half << 1);
      v2f a, bf;
      a.x = sT1[l * 32 + ka];
      a.y = sT1[l * 32 + ka + 1];
      bf.x = wp[0];
      bf.y = wp[32];
      c = wmma4(a, bf, c);
      wp += 128;
    }
    const float bb = me2[n0 + l];
#pragma unroll
    for (int r = 0; r < 8; ++r)
      eOut[(b * 256 + i * 16 + r + (half << 3)) * 32 + n0 + l] = c[r] + bb;
  }
}

// ---------------------------------------------------------------------------
// PNA pooling: g[b,0:64]=mean, 64:128=max, 128:192=min, 192:256=std(ddof=1);
// sum of node (16 samples) and edge (256 samples) statistics.
// ---------------------------------------------------------------------------
__global__ void pna_kernel(const float* __restrict__ nF,
                           const float* __restrict__ eF,
                           float* __restrict__ g) {
  const int b = blockIdx.x, d = threadIdx.x;  // 64 threads
  float s = 0.f, s2 = 0.f, mx = -3.4e38f, mn = 3.4e38f;
  for (int n = 0; n < 16; ++n) {
    const float x = nF[(b * 16 + n) * 64 + d];
    s += x; s2 += x * x; mx = fmaxf(mx, x); mn = fminf(mn, x);
  }
  const float meanN = s * (1.f / 16.f);
  const float varN = fmaxf((s2 - 16.f * meanN * meanN) * (1.f / 15.f), 0.f);
  float se = 0.f, se2 = 0.f, mxe = -3.4e38f, mne = 3.4e38f;
  for (int j = 0; j < 256; ++j) {
    const float x = eF[(b * 256 + j) * 64 + d];
    se += x; se2 += x * x; mxe = fmaxf(mxe, x); mne = fminf(mne, x);
  }
  const float meanE = se * (1.f / 256.f);
  const float varE = fmaxf((se2 - 256.f * meanE * meanE) * (1.f / 255.f), 0.f);
  g[b * 256 + d] = meanN + meanE;
  g[b * 256 + 64 + d] = mx + mxe;
  g[b * 256 + 128 + d] = mn + mne;
  g[b * 256 + 192 + d] = sqrtf(varN) + sqrtf(varE);
}

__global__ void reparam_kernel(const float* __restrict__ means,
                               const float* __restrict__ logv,
                               const float* __restrict__ eps,
                               float* __restrict__ z, int n) {
  const int i = blockIdx.x * 256 + threadIdx.x;
  if (i < n) z[i] = means[i] + eps[i] * __expf(0.5f * logv[i]);
}

// ---------------------------------------------------------------------------
// Host-side orchestration
// ---------------------------------------------------------------------------
static inline const float* F(void* const* din, int i) {
  return (const float*)din[i];
}

static inline void gemm(hipStream_t s, const float* A, const float* W,
                        const float* bias, float* C, int M, int N, int K,
                        int act, int aRowsPB = (1 << 30), long long aBatch = 0,
                        int aRowStride = -1, int aHead = 64) {
  if (aRowStride < 0) aRowStride = K;
  dim3 grid((((N + 15) / 16) + 3) / 4, M / 16, 1);
  if ((N & 15) == 0 && (M & 15) == 0)
    gemm_kernel<true><<<grid, 128, 0, s>>>(A, W, bias, C, M, N, K, aRowsPB,
                                           aBatch, aRowStride, aHead, act);
  else
    gemm_kernel<false><<<grid, 128, 0, s>>>(A, W, bias, C, M, N, K, aRowsPB,
                                            aBatch, aRowStride, aHead, act);
}

struct TfP {
  const float *Fb1, *Fb2, *Fw1, *Fw2, *Me1, *Me2, *Mn1, *Mn2, *We, *Wk, *Wn,
      *Wq, *Wv, *be, *bk, *bn, *bq, *bv, *me1, *me2, *mn1, *mn2;
};

static TfP tf_params(void* const* din, int base, int l) {
  TfP t;
  t.Fb1 = F(din, base + 0) + l * 512;
  t.Fb2 = F(din, base + 1) + l * 512;
  t.Fw1 = F(din, base + 2) + l * 16384;
  t.Fw2 = F(din, base + 3) + l * 16384;
  t.Me1 = F(din, base + 4) + l * 1024;
  t.Me2 = F(din, base + 5) + l * 1024;
  t.Mn1 = F(din, base + 6) + l * 4096;
  t.Mn2 = F(din, base + 7) + l * 4096;
  t.We = F(din, base + 8) + l * 16384;
  t.Wk = F(din, base + 9) + l * 32768;
  t.Wn = F(din, base + 10) + l * 32768;
  t.Wq = F(din, base + 11) + l * 32768;
  t.Wv = F(din, base + 12) + l * 32768;
  t.be = F(din, base + 13) + l * 32;
  t.bk = F(din, base + 14) + l * 512;
  t.bn = F(din, base + 15) + l * 64;
  t.bq = F(din, base + 16) + l * 512;
  t.bv = F(din, base + 17) + l * 512;
  t.me1 = F(din, base + 18) + l * 32;
  t.me2 = F(din, base + 19) + l * 32;
  t.mn1 = F(din, base + 20) + l * 64;
  t.mn2 = F(din, base + 21) + l * 64;
  return t;
}

static void tf_layer(const TfP& p, const float* nIn, const float* eIn,
                     float* nOut, float* eOut, float* q, float* k, float* v,
                     float* wv, float* nT1, float* nT2, hipStream_t s) {
  qkv_kernel<<<dim3(1, 128, 24), 128, 0, s>>>(nIn, p.Wq, p.Wk, p.Wv, p.bq,
                                              p.bk, p.bv, q, k, v);
  attn_kernel<<<2048, 512, 0, s>>>(q, k, v, eIn, p.Fw1, p.Fw2, p.Fb1, p.Fb2,
                                   p.We, p.be, p.Me1, p.me1, p.Me2, p.me2, wv,
                                   eOut);
  // node path: nn0 = relu(nodes_heads @ Wn + bn); mlp(Mn1,Mn2)
  gemm(s, wv, p.Wn, p.bn, nT1, 2048, 64, 512, 1, /*aRowsPB*/ 16,
       /*aBatch*/ 8192, /*aRowStride*/ 64, /*aHead*/ 1024);
  gemm(s, nT1, p.Mn1, p.mn1, nT2, 2048, 64, 64, 1);
  gemm(s, nT2, p.Mn2, p.mn2, nOut, 2048, 64, 64, 0);
}

extern "C" void kernel_launch(void* const* d_in, const int* in_sizes, int n_in,
                              void* d_out, int out_size, void* d_ws,
                              size_t ws_size, hipStream_t stream) {
  (void)in_sizes; (void)n_in; (void)out_size; (void)ws_size;
  const float* edgesIn = F(d_in, 0);
  const float* eps = F(d_in, 1);
  const float* nodesIn = F(d_in, 2);

  float* out = (float*)d_out;
  float* out_dn = out;              // [128,16,16]
  float* out_de = out + 32768;      // [128,16,16,8]
  float* out_mu = out + 98304;      // [128,1024]
  float* out_lv = out + 229376;     // [128,1024]

  // workspace carve (floats); total ~38 MB
  float* w = (float*)d_ws;
  float* nA = w;    w += 131072;    // [2048,64]
  float* nB = w;    w += 131072;
  float* nT1 = w;   w += 131072;
  float* nT2 = w;   w += 131072;
  float* eA = w;    w += 1048576;   // [32768,32]
  float* eB = w;    w += 1048576;
  float* qb = w;    w += 1048576;   // [B,H,16,64]
  float* kb = w;    w += 1048576;
  float* vb = w;    w += 1048576;
  float* wvb = w;   w += 1048576;
  float* eLin = w;  w += 2097152;   // [32768,64]
  float* gb = w;    w += 32768;     // [128,256]
  float* zb = w;    w += 131072;    // [128,1024]
  float* t64 = w;   w += 8192;      // [128,64]
  float* t512 = w;  w += 65536;     // [128,512]
  float* t6144 = w; w += 786432;    // [128,6144]

  // ---------------- encoder ----------------
  gemm(stream, nodesIn, F(d_in, 47), F(d_in, 49), nA, 2048, 64, 16, 1);  // inN1
  gemm(stream, nA, F(d_in, 48), F(d_in, 50), nB, 2048, 64, 64, 1);       // inN2
  gemm(stream, edgesIn, F(d_in, 43), F(d_in, 45), eA, 32768, 32, 8, 1);  // inE1
  gemm(stream, eA, F(d_in, 44), F(d_in, 46), eB, 32768, 32, 32, 1);      // inE2

  float* nCur = nB; float* nNxt = nA;
  float* eCur = eB; float* eNxt = eA;
  for (int l = 0; l < 4; ++l) {
    TfP p = tf_params(d_in, 55, l);
    tf_layer(p, nCur, eCur, nNxt, eNxt, qb, kb, vb, wvb, nT1, nT2, stream);
    float* t = nCur; nCur = nNxt; nNxt = t;
    t = eCur; eCur = eNxt; eNxt = t;
  }
  gemm(stream, eCur, F(d_in, 41), F(d_in, 42), eLin, 32768, 64, 32, 0);  // Wle
  pna_kernel<<<128, 64, 0, stream>>>(nCur, eLin, gb);
  gemm(stream, gb, F(d_in, 51), F(d_in, 53), t64, 128, 64, 256, 1);      // m1
  gemm(stream, t64, F(d_in, 52), F(d_in, 54), out_mu, 128, 1024, 64, 1); // m2
  gemm(stream, gb, F(d_in, 77), F(d_in, 79), t64, 128, 64, 256, 1);      // v1
  gemm(stream, t64, F(d_in, 78), F(d_in, 80), out_lv, 128, 1024, 64, 1); // v2
  reparam_kernel<<<512, 256, 0, stream>>>(out_mu, out_lv, eps, zb, 131072);

  // ---------------- decoder ----------------
  gemm(stream, zb, F(d_in, 7), F(d_in, 9), t512, 128, 512, 1024, 1);      // cn1
  gemm(stream, t512, F(d_in, 8), F(d_in, 10), nB, 128, 1024, 512, 1);     // cn2
  gemm(stream, zb, F(d_in, 3), F(d_in, 5), t6144, 128, 6144, 1024, 1);    // ce1
  gemm(stream, t6144, F(d_in, 4), F(d_in, 6), eB, 128, 8192, 6144, 1);    // ce2

  nCur = nB; nNxt = nA; eCur = eB; eNxt = eA;
  for (int l = 0; l < 4; ++l) {
    TfP p = tf_params(d_in, 19, l);
    tf_layer(p, nCur, eCur, nNxt, eNxt, qb, kb, vb, wvb, nT1, nT2, stream);
    float* t = nCur; nCur = nNxt; nNxt = t;
    t = eCur; eCur = eNxt; eNxt = t;
  }
  gemm(stream, nCur, F(d_in, 15), F(d_in, 17), nT1, 2048, 64, 64, 1);     // on1
  gemm(stream, nT1, F(d_in, 16), F(d_in, 18), out_dn, 2048, 16, 64, 2);   // on2 (+sigmoid col0)
  gemm(stream, eCur, F(d_in, 11), F(d_in, 13), eNxt, 32768, 32, 32, 1);   // oe1
  gemm(stream, eNxt, F(d_in, 12), F(d_in, 14), out_de, 32768, 8, 32, 0);  // oe2
}